// RetrievalModel_16217796510376
// MI455X (gfx1250) — compile-verified
//
#include <hip/hip_runtime.h>
#include <math.h>

typedef __attribute__((ext_vector_type(2))) float v2f;
typedef __attribute__((ext_vector_type(8))) float v8f;

#define Bsz 128
#define Ssz 512
#define Dsz 1024
#define Psz 32
#define BETAc 0.4f
#define TEMPc 0.1f

// D = A(16x4) * B(4x16) + C  via CDNA5 f32 WMMA
__device__ __forceinline__ v8f wmma4(v2f a, v2f b, v8f c) {
  return __builtin_amdgcn_wmma_f32_16x16x4_f32(false, a, false, b, (short)0, c,
                                               false, false);
}

// ---------------------------------------------------------------------------
// Generic WMMA GEMM: out = act( A @ W (+ bias[*rowScale]) (+ res) )
//   A: [M,lda]; W row-major [K,N] (TRANSW=0) or element (col,red)=W[col*ldw+red]
//   ACT: 0=none, 1=sigmoid, 2=leaky_relu(0.01)
//   BIASMODE: 0=none, 1=bias, 2=bias*rowScale ; RESMODE: 0/1
// Block = 128 threads (4 waves); tile = 16 (M) x 64 (N), one 16x16 per wave.
// Compile-time flags -> branch-free inner loop; pointer-increment addressing.
// ---------------------------------------------------------------------------
template <int TRANSW, int ACT, int BIASMODE, int RESMODE>
__global__ __launch_bounds__(128)
void gemm_wmma_kernel(const float* __restrict__ A, int lda,
                      const float* __restrict__ W, int ldw,
                      const float* __restrict__ bias,
                      const float* __restrict__ rowScale,
                      const float* __restrict__ res, int ldr,
                      float* __restrict__ out, int ldo,
                      int M, int N, int K) {
  int lane = threadIdx.x & 31;
  int wvid = threadIdx.x >> 5;
  int m0 = blockIdx.y * 16;
  int n0 = blockIdx.x * 64 + wvid * 16;
  if (m0 >= M || n0 >= N) return;  // wave-uniform -> EXEC stays all-ones
  int hh = lane >> 4, l16 = lane & 15, kb = hh * 2;
  v8f c = {0.f, 0.f, 0.f, 0.f, 0.f, 0.f, 0.f, 0.f};

  const float* ap = A + (size_t)(m0 + l16) * lda + kb;
  const float* wp0;
  size_t wstep;
  if (TRANSW) {
    wp0 = W + (size_t)(n0 + l16) * ldw + kb;
    wstep = 4;
  } else {
    wp0 = W + (size_t)kb * ldw + (n0 + l16);
    wstep = 4 * (size_t)ldw;
  }
#pragma unroll 8
  for (int k = 0; k < K; k += 4) {
    v2f a = *(const v2f*)ap;
    v2f b;
    if (TRANSW) {
      b = *(const v2f*)wp0;
    } else {
      b.x = wp0[0];
      b.y = wp0[ldw];
    }
    c = wmma4(a, b, c);
    ap += 4;
    wp0 += wstep;
  }
  for (int v = 0; v < 8; ++v) {
    int row = m0 + v + 8 * hh;
    int col = n0 + l16;
    float x = c[v];
    if (BIASMODE == 1) x += bias[col];
    if (BIASMODE == 2) x += bias[col] * rowScale[row];
    if (RESMODE) x += res[(size_t)row * ldr + col];
    if (ACT == 1) x = 1.f / (1.f + __expf(-x));
    else if (ACT == 2) x = (x > 0.f) ? x : 0.01f * x;
    out[(size_t)row * ldo + col] = x;
  }
}

// ---------------------------------------------------------------------------
// Fused score kernel. For 64 rows m = b*S+s:
//   score[m] = x_tail[m].qvec[b] + lrelu(feat[m]@wf1 + bf1).w2q[b] + constd[b]
// feat = [x_tail | x_rel] (K=2048). H tiles never leave registers.
// A chunks staged to LDS with CDNA5 async copies (global_load_async_to_lds,
// ASYNCcnt); block = 256 thr (8 waves = 4 mi x 2 ni); N_CHUNK=32, K_CHUNK=128.
// ---------------------------------------------------------------------------
__global__ __launch_bounds__(256)
void score_kernel(const float* __restrict__ x_tail, const float* __restrict__ x_rel,
                  const float* __restrict__ x_mask,
                  const float* __restrict__ wf1, const float* __restrict__ bf1,
                  const float* __restrict__ qvec, const float* __restrict__ w2q,
                  const float* __restrict__ constd, float* __restrict__ scores) {
  __shared__ float ldsA[64][132];  // 64 rows x 128 K, +4 pad (row = 528B = 33*16)
  __shared__ float accp[2][64];    // per-ni partial h-dot accumulators
  __shared__ float tdot[64];       // x_tail . qvec
  int tid = threadIdx.x;
  int m0 = blockIdx.x * 64;
  int b = m0 >> 9;  // / Ssz (64 | 512 so whole tile shares b)
  int lane = tid & 31, wvid = tid >> 5;
  int mi = wvid >> 1;  // 0..3
  int ni = wvid & 1;   // 0..1
  int hh = lane >> 4, l16 = lane & 15, kb = hh * 2;

  if (tid < 128) accp[tid >> 6][tid & 63] = 0.f;
  if (tid < 64) {
    const float* tr = x_tail + (size_t)(m0 + tid) * Dsz;
    const float* qv = qvec + (size_t)b * Dsz;
    float s = 0.f;
    for (int k = 0; k < Dsz; k += 4) {
      float4 t4 = *(const float4*)(tr + k);
      float4 q4 = *(const float4*)(qv + k);
      s += t4.x * q4.x + t4.y * q4.y + t4.z * q4.z + t4.w * q4.w;
    }
    tdot[tid] = s;
  }
  __syncthreads();

  const float* w2 = w2q + (size_t)b * Dsz;
  for (int nc = 0; nc < Dsz; nc += 32) {
    int n = nc + ni * 16 + l16;
    v8f c = {0.f, 0.f, 0.f, 0.f, 0.f, 0.f, 0.f, 0.f};
    for (int kc = 0; kc < 2 * Dsz; kc += 128) {
      __syncthreads();  // all waves done reading previous chunk
      const float* base = (kc < Dsz) ? x_tail : x_rel;
      int koff = (kc < Dsz) ? kc : (kc - Dsz);
      // 2048 x 16B async copies: global -> LDS, tracked by ASYNCcnt
#pragma unroll
      for (int i = 0; i < 8; ++i) {
        int fidx = tid + 256 * i;  // 0..2047
        int rr = fidx >> 5;        // row 0..63
        int c4 = fidx & 31;        // float4 within 128 cols
        unsigned ldsoff = (unsigned)(size_t)&ldsA[rr][c4 * 4];
        const float* g = base + (size_t)(m0 + rr) * Dsz + koff + c4 * 4;
        asm volatile("global_load_async_to_lds_b128 %0, %1, off"
                     :: "v"(ldsoff), "v"(g) : "memory");
      }
      asm volatile("s_wait_asynccnt 0" ::: "memory");
      __syncthreads();
      const float* arow = &ldsA[mi * 16 + l16][kb];
      const float* wr = wf1 + (size_t)(kc + kb) * Dsz + n;
#pragma unroll 8
      for (int kk = 0; kk < 128; kk += 4) {
        v2f a = *(const v2f*)arow;
        v2f bb;
        bb.x = wr[0];
        bb.y = wr[Dsz];
        c = wmma4(a, bb, c);
        arow += 4;
        wr += 4 * (size_t)Dsz;
      }
    }
    // epilogue: lrelu(H + bf1) . w2q over this 16-col slab
    float wcol = w2[n];
    float bcol = bf1[n];
#pragma unroll
    for (int v = 0; v < 8; ++v) {
      float h = c[v] + bcol;
      h = (h > 0.f) ? h : 0.01f * h;
      float p = h * wcol;
      for (int msk = 8; msk >= 1; msk >>= 1) p += __shfl_xor(p, msk, 32);
      if (l16 == 0) accp[ni][mi * 16 + v + 8 * hh] += p;  // unique writer
    }
  }
  __syncthreads();
  if (tid < 64) {
    int m = m0 + tid;
    int s = m & (Ssz - 1);
    float sc = tdot[tid] + accp[0][tid] + accp[1][tid] + constd[b];
    bool valid = x_mask[(size_t)b * Ssz + s] > 0.01f;
    scores[m] = valid ? sc : -INFINITY;
  }
}

// u = (BETA*iq + (1-BETA)*hq) / (sqrt(D)*TEMP)
__global__ void u_kernel(const float* __restrict__ iq, const float* __restrict__ hq,
                         float* __restrict__ u, int n) {
  int i = blockIdx.x * 256 + threadIdx.x;
  if (i < n) u[i] = (BETAc * iq[i] + (1.f - BETAc) * hq[i]) * (0.03125f / TEMPc);
}

// constd[b] = bf2.qvec[b] + bk.u[b]
__global__ __launch_bounds__(256)
void constd_kernel(const float* __restrict__ bf2, const float* __restrict__ qvec,
                   const float* __restrict__ bk, const float* __restrict__ u,
                   float* __restrict__ constd) {
  __shared__ float red[256];
  int b = blockIdx.x;
  float s = 0.f;
  for (int k = threadIdx.x; k < Dsz; k += 256)
    s += bf2[k] * qvec[(size_t)b * Dsz + k] + bk[k] * u[(size_t)b * Dsz + k];
  red[threadIdx.x] = s;
  __syncthreads();
  for (int st = 128; st > 0; st >>= 1) {
    if (threadIdx.x < st) red[threadIdx.x] += red[threadIdx.x + st];
    __syncthreads();
  }
  if (threadIdx.x == 0) constd[b] = red[0];
}

// softmax over S then iterative top-32 argmax (ties -> lowest index, like jax)
__global__ __launch_bounds__(32)
void topk_kernel(const float* __restrict__ scores, float* __restrict__ top_w,
                 int* __restrict__ top_idx) {
  int b = blockIdx.x;
  int lane = threadIdx.x;
  float v[16];
  float mx = -INFINITY;
  for (int j = 0; j < 16; ++j) {
    v[j] = scores[(size_t)b * Ssz + lane + 32 * j];
    mx = fmaxf(mx, v[j]);
  }
  for (int m = 16; m >= 1; m >>= 1) mx = fmaxf(mx, __shfl_xor(mx, m, 32));
  float sum = 0.f;
  for (int j = 0; j < 16; ++j) {
    v[j] = __expf(v[j] - mx);  // exp(-inf)=0 for masked
    sum += v[j];
  }
  for (int m = 16; m >= 1; m >>= 1) sum += __shfl_xor(sum, m, 32);
  float inv = 1.f / sum;
  for (int p = 0; p < Psz; ++p) {
    float bv = -1.f;
    int bi = 0x7fffffff;
    for (int j = 0; j < 16; ++j)
      if (v[j] > bv) { bv = v[j]; bi = lane + 32 * j; }
    for (int m = 16; m >= 1; m >>= 1) {
      float ov = __shfl_xor(bv, m, 32);
      int oi = __shfl_xor(bi, m, 32);
      if (ov > bv || (ov == bv && oi < bi)) { bv = ov; bi = oi; }
    }
    if (lane == (bi & 31)) v[bi >> 5] = -1.f;  // remove winner
    if (lane == 0) {
      top_w[b * Psz + p] = bv * inv;
      top_idx[b * Psz + p] = bi;
    }
  }
}

// Asel[j] = [x_tail[b,s] | x_rel[b,s]] for selected s
__global__ __launch_bounds__(256)
void gather_kernel(const float* __restrict__ x_tail, const float* __restrict__ x_rel,
                   const int* __restrict__ top_idx, float* __restrict__ Asel) {
  int j = blockIdx.x;
  int b = j >> 5;  // / Psz
  int s = top_idx[j];
  const float* t = x_tail + ((size_t)b * Ssz + s) * Dsz;
  const float* r = x_rel + ((size_t)b * Ssz + s) * Dsz;
  float* o = Asel + (size_t)j * (2 * Dsz);
  for (int k = threadIdx.x * 4; k < Dsz; k += 256 * 4) {
    *(float4*)(o + k) = *(const float4*)(t + k);
    *(float4*)(o + Dsz + k) = *(const float4*)(r + k);
  }
}

// cR[b] = sum_p w[b,p]*Rsel[b*P+p];  asum[b] = sum_p w[b,p]
__global__ __launch_bounds__(256)
void wreduce_kernel(const float* __restrict__ Rsel, const float* __restrict__ top_w,
                    float* __restrict__ cR, float* __restrict__ asum) {
  int b = blockIdx.y;
  int k = blockIdx.x * 256 + threadIdx.x;
  float s = 0.f;
  for (int p = 0; p < Psz; ++p)
    s += top_w[b * Psz + p] * Rsel[(size_t)(b * Psz + p) * Dsz + k];
  cR[(size_t)b * Dsz + k] = s;
  if (blockIdx.x == 0 && threadIdx.x == 0) {
    float a = 0.f;
    for (int p = 0; p < Psz; ++p) a += top_w[b * Psz + p];
    asum[b] = a;
  }
}

__global__ void combine_kernel(const float* __restrict__ trout,
                               const float* __restrict__ tout,
                               const float* __restrict__ tg,
                               const float* __restrict__ iout,
                               const float* __restrict__ ig,
                               float* __restrict__ out, int n) {
  int i = blockIdx.x * 256 + threadIdx.x;
  if (i < n) out[i] = trout[i] + tout[i] * tg[i] + iout[i] * ig[i];
}

#define GEMM(TR_, ACT_, BM_, RM_, A_, lda_, W_, ldw_, bias_, rs_, res_, ldr_, out_, ldo_, M_, N_, K_) \
  gemm_wmma_kernel<TR_, ACT_, BM_, RM_>                                                                \
      <<<dim3((N_) / 64, (M_) / 16), dim3(128), 0, stream>>>(                                          \
          A_, lda_, W_, ldw_, bias_, rs_, res_, ldr_, out_, ldo_, M_, N_, K_)

extern "C" void kernel_launch(void* const* d_in, const int* in_sizes, int n_in,
                              void* d_out, int out_size, void* d_ws, size_t ws_size,
                              hipStream_t stream) {
  (void)in_sizes; (void)n_in; (void)out_size; (void)ws_size;
  const float* x_head = (const float*)d_in[0];
  const float* x_rel  = (const float*)d_in[1];
  const float* x_tail = (const float*)d_in[2];
  const float* x_mask = (const float*)d_in[3];
  const float* x_img  = (const float*)d_in[4];
  const float* wf1 = (const float*)d_in[5];  const float* bf1 = (const float*)d_in[6];
  const float* wf2 = (const float*)d_in[7];  const float* bf2 = (const float*)d_in[8];
  const float* wk  = (const float*)d_in[9];  const float* bk  = (const float*)d_in[10];
  const float* wvw = (const float*)d_in[11]; const float* bvv = (const float*)d_in[12];
  const float* wp  = (const float*)d_in[13]; const float* bp  = (const float*)d_in[14];
  const float* wiq = (const float*)d_in[15]; const float* biq = (const float*)d_in[16];
  const float* whq = (const float*)d_in[17]; const float* bhq = (const float*)d_in[18];
  const float* wlt = (const float*)d_in[19]; const float* blt = (const float*)d_in[20];
  const float* wli = (const float*)d_in[21]; const float* bli = (const float*)d_in[22];
  const float* wltr= (const float*)d_in[23]; const float* bltr= (const float*)d_in[24];
  const float* wgt = (const float*)d_in[25]; const float* bgt = (const float*)d_in[26];
  const float* wgi = (const float*)d_in[27]; const float* bgi = (const float*)d_in[28];
  float* out = (float*)d_out;

  const size_t BD = (size_t)Bsz * Dsz;  // 131072
  const int SEL = Bsz * Psz;            // 4096
  float* ws = (float*)d_ws;
  size_t o = 0;
  float* iq_    = ws + o; o += BD;
  float* hq_    = ws + o; o += BD;
  float* u_     = ws + o; o += BD;
  float* qvec_  = ws + o; o += BD;
  float* w2q_   = ws + o; o += BD;
  float* constd_= ws + o; o += 128;
  float* scores_= ws + o; o += (size_t)Bsz * Ssz;
  float* topw_  = ws + o; o += SEL;
  int*   topidx_= (int*)(ws + o); o += SEL;
  float* asum_  = ws + o; o += 128;
  float* Asel_  = ws + o; o += (size_t)SEL * 2 * Dsz;
  float* Hsel_  = ws + o; o += (size_t)SEL * Dsz;
  float* Rsel_  = ws + o; o += (size_t)SEL * Dsz;
  float* cR_    = ws + o; o += BD;
  float* ctx_   = ws + o; o += BD;
  float* xtr_   = ws + o; o += BD;
  float* tout_  = ws + o; o += BD;
  float* iout_  = ws + o; o += BD;
  float* trout_ = ws + o; o += BD;
  float* tg_    = ws + o; o += BD;
  float* ig_    = ws + o; o += BD;

  // queries
  GEMM(0, 0, 1, 0, x_img,  Dsz, wiq, Dsz, biq, nullptr, nullptr, 0, iq_, Dsz, Bsz, Dsz, Dsz);
  GEMM(0, 0, 1, 0, x_head, Dsz, whq, Dsz, bhq, nullptr, nullptr, 0, hq_, Dsz, Bsz, Dsz, Dsz);
  u_kernel<<<dim3((int)(BD / 256)), dim3(256), 0, stream>>>(iq_, hq_, u_, (int)BD);
  // fold wk, wf2 into per-batch vectors (transposed tiny GEMMs)
  GEMM(1, 0, 0, 0, u_,    Dsz, wk,  Dsz, nullptr, nullptr, nullptr, 0, qvec_, Dsz, Bsz, Dsz, Dsz);
  GEMM(1, 0, 0, 0, qvec_, Dsz, wf2, Dsz, nullptr, nullptr, nullptr, 0, w2q_,  Dsz, Bsz, Dsz, Dsz);
  constd_kernel<<<dim3(Bsz), dim3(256), 0, stream>>>(bf2, qvec_, bk, u_, constd_);
  // heavy fused pass: scores for all B*S rows (K and V never materialized)
  score_kernel<<<dim3((Bsz * Ssz) / 64), dim3(256), 0, stream>>>(
      x_tail, x_rel, x_mask, wf1, bf1, qvec_, w2q_, constd_, scores_);
  // softmax + top-P
  topk_kernel<<<dim3(Bsz), dim3(32), 0, stream>>>(scores_, topw_, topidx_);
  // recompute R only for the 4096 selected rows
  gather_kernel<<<dim3(SEL), dim3(256), 0, stream>>>(x_tail, x_rel, topidx_, Asel_);
  GEMM(0, 2, 1, 0, Asel_, 2 * Dsz, wf1, Dsz, bf1, nullptr, nullptr, 0, Hsel_, Dsz, SEL, Dsz, 2 * Dsz);
  GEMM(0, 0, 1, 1, Hsel_, Dsz, wf2, Dsz, bf2, nullptr, Asel_, 2 * Dsz, Rsel_, Dsz, SEL, Dsz, Dsz);
  wreduce_kernel<<<dim3(Dsz / 256, Bsz), dim3(256), 0, stream>>>(Rsel_, topw_, cR_, asum_);
  // ctx = cR@wv + asum*bv ; x_triple = ctx@wp + bp
  GEMM(0, 0, 2, 0, cR_,  Dsz, wvw, Dsz, bvv, asum_,   nullptr, 0, ctx_, Dsz, Bsz, Dsz, Dsz);
  GEMM(0, 0, 1, 0, ctx_, Dsz, wp,  Dsz, bp,  nullptr, nullptr, 0, xtr_, Dsz, Bsz, Dsz, Dsz);
  // gated projection combine
  GEMM(0, 0, 1, 0, x_head, Dsz, wlt,  Dsz, blt,  nullptr, nullptr, 0, tout_,  Dsz, Bsz, Dsz, Dsz);
  GEMM(0, 0, 1, 0, x_img,  Dsz, wli,  Dsz, bli,  nullptr, nullptr, 0, iout_,  Dsz, Bsz, Dsz, Dsz);
  GEMM(0, 0, 1, 0, xtr_,   Dsz, wltr, Dsz, bltr, nullptr, nullptr, 0, trout_, Dsz, Bsz, Dsz, Dsz);
  GEMM(0, 1, 1, 0, x_head, Dsz, wgt,  Dsz, bgt,  nullptr, nullptr, 0, tg_,    Dsz, Bsz, Dsz, Dsz);
  GEMM(0, 1, 1, 0, x_img,  Dsz, wgi,  Dsz, bgi,  nullptr, nullptr, 0, ig_,    Dsz, Bsz, Dsz, Dsz);
  combine_kernel<<<dim3((int)(BD / 256)), dim3(256), 0, stream>>>(
      trout_, tout_, tg_, iout_, ig_, out, (int)BD);
}